// MMLDA_layer_82858509075021
// MI455X (gfx1250) — compile-verified
//
#include <hip/hip_runtime.h>
#include <hip/hip_bf16.h>
#include <math.h>

typedef __attribute__((ext_vector_type(2))) float v2f;
typedef __attribute__((ext_vector_type(8))) float v8f;
typedef __attribute__((__vector_size__(16))) int v4i;

#define P 512          // feature dim (fixed by reference)
#define BM 128         // rows per block tile
#define BN 64          // cols per block tile
#define KC 32          // K chunk
#define LDST 36        // LDS row stride in dwords: pad 32->36 (16B aligned, bank-conflict free)
#define EPS 1e-10f

#if __has_builtin(__builtin_amdgcn_global_load_async_to_lds_b128) && \
    __has_builtin(__builtin_amdgcn_s_wait_asynccnt)
#define USE_ASYNC 1
// memory -> LDS DMA, no VGPR round trip, tracked on ASYNCcnt.
// Builtin signature (from hipcc diagnostic): (v4i addrspace(1)*, v4i addrspace(3)*, imm offset, imm cpol)
#define ASYNC_CP16(g, l)                                                      \
    __builtin_amdgcn_global_load_async_to_lds_b128(                           \
        (__attribute__((address_space(1))) v4i*)(g),                          \
        (__attribute__((address_space(3))) v4i*)(l), 0, 0)
#else
#define USE_ASYNC 0
#endif

// ---------------- Kernel A / B: wave-per-row sum of squares ----------------

__global__ __launch_bounds__(256) void row_sumsq_kernel(
    const float* __restrict__ m, float* __restrict__ out_sq, int nrows)
{
    const int row  = blockIdx.x * 8 + (threadIdx.x >> 5);
    const int lane = threadIdx.x & 31;
    if (row >= nrows) return;
    const float* p = m + (size_t)row * P;
    float s = 0.f;
    #pragma unroll
    for (int i = 0; i < P / 32; ++i) {
        float v = p[lane + 32 * i];
        s = fmaf(v, v, s);
    }
    #pragma unroll
    for (int d = 16; d; d >>= 1) s += __shfl_xor(s, d, 32);
    if (lane == 0) out_sq[row] = s;
}

__global__ __launch_bounds__(256) void row_scale_kernel(
    const float* __restrict__ x, const float* __restrict__ msq,
    float* __restrict__ s_row, float* __restrict__ xsq_row, int nrows)
{
    const int row  = blockIdx.x * 8 + (threadIdx.x >> 5);
    const int lane = threadIdx.x & 31;
    if (row >= nrows) return;
    const float* p = x + (size_t)row * P;
    float s = 0.f;
    #pragma unroll
    for (int i = 0; i < P / 32; ++i) {
        float v = p[lane + 32 * i];
        s = fmaf(v, v, s);
    }
    #pragma unroll
    for (int d = 16; d; d >>= 1) s += __shfl_xor(s, d, 32);
    if (lane == 0) {
        float scale = sqrtf(msq[0]);         // ||means[0]||
        float nrm   = sqrtf(s);
        float sb    = scale / (nrm + EPS);   // per-row rescale factor
        s_row[row]   = sb;
        xsq_row[row] = sb * sb * s;          // ||xn_row||^2 exactly as reference computes it
    }
}

// ---------------- Kernel C: f32 WMMA GEMM + fused epilogue ----------------
// out[b][c] = 2*s_b*(x_b . mu_c) - xsq_b - msq_c

__global__ __launch_bounds__(256) void gemm_logits_kernel(
    const float* __restrict__ x, const float* __restrict__ means,
    const float* __restrict__ s_row, const float* __restrict__ xsq_row,
    const float* __restrict__ msq, float* __restrict__ out, int ncls)
{
    __shared__ float lx[2][BM * LDST];   // x tile     128 x 32 (+pad)
    __shared__ float lm[2][BN * LDST];   // means tile  64 x 32 (+pad)
    __shared__ float sS[BM], sX[BM], sM[BN];

    const int tid  = threadIdx.x;
    const int lane = tid & 31;
    const int wave = tid >> 5;            // 8 waves
    const int wr   = wave & 3;            // 4 row-groups of 32 rows
    const int wc   = wave >> 2;           // 2 col-groups of 32 cols
    const int row0 = blockIdx.x * BM;
    const int col0 = blockIdx.y * BN;

    // Stage epilogue vectors.
    if (tid < BM) { sS[tid] = s_row[row0 + tid]; sX[tid] = xsq_row[row0 + tid]; }
    if (tid < BN) { int c = col0 + tid; sM[tid] = (c < ncls) ? msq[c] : 0.f; }

    // One-time zero-fill of means-tile rows that are out of range (async loads
    // never touch these rows, so there is no DS-vs-async write race; cols >= ncls
    // are masked on store, this just keeps the accumulators finite).
    {
        const int r  = tid >> 2;          // 0..63
        const int ko = (tid & 3) * 8;
        if (col0 + r >= ncls) {
            float4 z = {0.f, 0.f, 0.f, 0.f};
            #pragma unroll
            for (int b = 0; b < 2; ++b) {
                float4* d = (float4*)(&lm[b][r * LDST + ko]);
                d[0] = z; d[1] = z;
            }
        }
    }

    auto loadChunk = [&](int kc, int buf) {
        const int kb = kc * KC;
        {   // x: 128 rows x 32 floats; 2 threads/row, 4x 16B each
            const int r  = tid >> 1;
            const int ko = (tid & 1) * 16;
            const float* g = x + (size_t)(row0 + r) * P + kb + ko;
            float* l = &lx[buf][r * LDST + ko];
#if USE_ASYNC
            ASYNC_CP16(g + 0, l + 0);
            ASYNC_CP16(g + 4, l + 4);
            ASYNC_CP16(g + 8, l + 8);
            ASYNC_CP16(g + 12, l + 12);
#else
            const float4* gv = (const float4*)g;
            float4* d = (float4*)l;
            d[0] = gv[0]; d[1] = gv[1]; d[2] = gv[2]; d[3] = gv[3];
#endif
        }
        {   // means: 64 rows x 32 floats; 4 threads/row, 2x 16B each
            const int r  = tid >> 2;
            const int ko = (tid & 3) * 8;
            const int c  = col0 + r;
            if (c < ncls) {
                const float* g = means + (size_t)c * P + kb + ko;
                float* l = &lm[buf][r * LDST + ko];
#if USE_ASYNC
                ASYNC_CP16(g + 0, l + 0);
                ASYNC_CP16(g + 4, l + 4);
#else
                const float4* gv = (const float4*)g;
                float4* d = (float4*)l;
                d[0] = gv[0]; d[1] = gv[1];
#endif
            }
        }
    };

    v8f acc[2][2] = {};

    // Fragment addressing per V_WMMA_F32_16X16X4_F32 layout:
    // A (16x4): lane&15 = M row, (lane>>4)*2 = first K of the pair held in 2 VGPRs.
    // B (4x16): mirrored (lane&15 = N col of means^T = means row), same K pairing.
    const int mrow = lane & 15;
    const int kq   = (lane >> 4) * 2;

    loadChunk(0, 0);
#if USE_ASYNC
    __builtin_amdgcn_s_wait_asynccnt(0);
#endif
    __syncthreads();

    const int NKC = P / KC;   // 16
    for (int kc = 0; kc < NKC; ++kc) {
        const int cur = kc & 1;
        if (kc + 1 < NKC) loadChunk(kc + 1, cur ^ 1);   // async prefetch overlaps compute

        const float* px0 = &lx[cur][(wr * 32 +      mrow) * LDST];
        const float* px1 = &lx[cur][(wr * 32 + 16 + mrow) * LDST];
        const float* pm0 = &lm[cur][(wc * 32 +      mrow) * LDST];
        const float* pm1 = &lm[cur][(wc * 32 + 16 + mrow) * LDST];

        #pragma unroll
        for (int kk = 0; kk < KC / 4; ++kk) {
            const int o = kk * 4 + kq;
            v2f a0 = *(const v2f*)(px0 + o);
            v2f a1 = *(const v2f*)(px1 + o);
            v2f b0 = *(const v2f*)(pm0 + o);
            v2f b1 = *(const v2f*)(pm1 + o);
            acc[0][0] = __builtin_amdgcn_wmma_f32_16x16x4_f32(false, a0, false, b0, (short)0, acc[0][0], false, false);
            acc[0][1] = __builtin_amdgcn_wmma_f32_16x16x4_f32(false, a0, false, b1, (short)0, acc[0][1], false, false);
            acc[1][0] = __builtin_amdgcn_wmma_f32_16x16x4_f32(false, a1, false, b0, (short)0, acc[1][0], false, false);
            acc[1][1] = __builtin_amdgcn_wmma_f32_16x16x4_f32(false, a1, false, b1, (short)0, acc[1][1], false, false);
        }
#if USE_ASYNC
        __builtin_amdgcn_s_wait_asynccnt(0);  // my async writes into next buffer done
#endif
        __syncthreads();                       // everyone's writes visible
    }

    // Epilogue: C/D layout -> vgpr i, lanes 0-15: M=i, N=lane; lanes 16-31: M=i+8, N=lane-16.
    const int nhalf = (lane >> 4) * 8;
    const int ncol  = lane & 15;
    #pragma unroll
    for (int rt = 0; rt < 2; ++rt) {
        #pragma unroll
        for (int ct = 0; ct < 2; ++ct) {
            const int rbase = wr * 32 + rt * 16;
            const int cbase = wc * 32 + ct * 16;
            const int c = col0 + cbase + ncol;
            if (c < ncls) {
                const float mq = sM[cbase + ncol];
                #pragma unroll
                for (int i = 0; i < 8; ++i) {
                    const int rl = rbase + i + nhalf;
                    float v = 2.f * sS[rl] * acc[rt][ct][i] - sX[rl] - mq;
                    out[(size_t)(row0 + rl) * ncls + c] = v;
                }
            }
        }
    }
}

// ---------------- Launch ----------------

extern "C" void kernel_launch(void* const* d_in, const int* in_sizes, int n_in,
                              void* d_out, int out_size, void* d_ws, size_t ws_size,
                              hipStream_t stream) {
    const float* x     = (const float*)d_in[0];
    const float* means = (const float*)d_in[1];
    float* out = (float*)d_out;

    const int B = in_sizes[0] / P;   // 8192
    const int C = in_sizes[1] / P;   // 1000

    float* ws   = (float*)d_ws;
    float* msq  = ws;                 // C floats (<=1024 slot)
    float* srow = ws + 1024;          // B floats
    float* xsq  = ws + 1024 + B;      // B floats

    row_sumsq_kernel<<<(C + 7) / 8, 256, 0, stream>>>(means, msq, C);
    row_scale_kernel<<<(B + 7) / 8, 256, 0, stream>>>(x, msq, srow, xsq, B);

    dim3 grid(B / BM, (C + BN - 1) / BN);   // 64 x 16
    gemm_logits_kernel<<<grid, 256, 0, stream>>>(x, means, srow, xsq, msq, out, C);
}